// GAT_61795989455227
// MI455X (gfx1250) — compile-verified
//
#include <hip/hip_runtime.h>
#include <hip/hip_bf16.h>
#include <math.h>

#define NNODES 50000
#define NEDGES 800000
#define ETOT   (NEDGES + NNODES)   // 850000 with self-loops
#define FIN    256
#define H1N    8
#define C1     64
#define F1     512                  // H1N*C1
#define FOUT   64
#define NSLOPE 0.2f

typedef float v2f __attribute__((ext_vector_type(2)));
typedef float v8f __attribute__((ext_vector_type(8)));

// ---- CDNA5 async memory->LDS path (guarded; falls back to float4 staging) ----
#if defined(__HIP_DEVICE_COMPILE__) && \
    __has_builtin(__builtin_amdgcn_global_load_async_to_lds_b128) && \
    __has_builtin(__builtin_amdgcn_s_wait_asynccnt)
#define USE_ASYNC_LDS 1
typedef int v4i __attribute__((vector_size(16)));
typedef __attribute__((address_space(1))) v4i as1_v4i;   // global
typedef __attribute__((address_space(3))) v4i as3_v4i;   // LDS
__device__ __forceinline__ void async_copy16(const float* g, float* l) {
    __builtin_amdgcn_global_load_async_to_lds_b128((as1_v4i*)g, (as3_v4i*)l, 0, 0);
}
#endif

// ---------------- helpers ----------------

__device__ __forceinline__ void edge_sd(const int* __restrict__ EI, int e, int& s, int& d) {
    if (e < NEDGES) { s = EI[e]; d = EI[NEDGES + e]; }
    else            { s = e - NEDGES; d = s; }
}

// float atomic max via int-max / uint-min bit trick (native L2 atomics, no CAS loop)
__device__ __forceinline__ void atomicMaxF(float* addr, float val) {
    if (val >= 0.0f) atomicMax((int*)addr, __float_as_int(val));
    else             atomicMin((unsigned int*)addr, __float_as_uint(val));
}

__global__ void k_fill(float* __restrict__ p, float v, int n) {
    int i = blockIdx.x * blockDim.x + threadIdx.x;
    if (i < n) p[i] = v;
}

// ---------------- GEMM 1: h1 = x @ W1   [50000x256]x[256x512] ----------------
// Block tile M=80 x N=128: 8 waves, each wave = 5 M-tiles x 1 N-tile, so each
// B fragment is loaded once and feeds 5 WMMAs (A fragments from LDS).
// LDS row stride 260 (260 % 64 == 4) -> conflict-free b64 A-fragment reads.
// Grid: (50000/80 = 625) x (512/128 = 4).

__global__ __launch_bounds__(256) void k_gemm1(const float* __restrict__ X,
                                               const float* __restrict__ W,
                                               float* __restrict__ H) {
    __shared__ float xs[80 * 260];
    const int m0  = blockIdx.x * 80;
    const int tid = threadIdx.x;
    // stage X[80][256] -> LDS (5120 float4 over 256 threads)
#ifdef USE_ASYNC_LDS
#pragma unroll
    for (int i = 0; i < 20; ++i) {
        int idx = tid + i * 256;
        int row = idx >> 6;                       // 64 float4 per 256-float row
        int k4  = idx & 63;
        async_copy16(X + (size_t)(m0 + row) * FIN + k4 * 4, &xs[row * 260 + k4 * 4]);
    }
    __builtin_amdgcn_s_wait_asynccnt(0);
#else
    const float4* x4 = (const float4*)(X + (size_t)m0 * FIN);
#pragma unroll
    for (int i = 0; i < 20; ++i) {
        int idx = tid + i * 256;
        int row = idx >> 6;
        int k4  = idx & 63;
        float4 v = x4[row * 64 + k4];
        float* dp = &xs[row * 260 + k4 * 4];
        dp[0] = v.x; dp[1] = v.y; dp[2] = v.z; dp[3] = v.w;
    }
#endif
    __syncthreads();

    const int lane = tid & 31;
    const int wave = tid >> 5;
    const int n0   = blockIdx.y * 128 + wave * 16;
    const int lm   = lane & 15;        // M (for A) / N (for B,C)
    const int lk   = lane >> 4;        // K-group select
    v8f acc[5] = {};
    for (int k0 = 0; k0 < FIN; k0 += 4) {
        const int ka = k0 + 2 * lk;    // ISA layout: VGPR j, group g -> K = j + 2g
        v2f b;                          // one B fragment, reused by 5 WMMAs
        b.x = W[(size_t)ka * F1 + n0 + lm];
        b.y = W[(size_t)(ka + 1) * F1 + n0 + lm];
#pragma unroll
        for (int mt = 0; mt < 5; ++mt) {
            v2f a;
            a.x = xs[(mt * 16 + lm) * 260 + ka];
            a.y = xs[(mt * 16 + lm) * 260 + ka + 1];
            acc[mt] = __builtin_amdgcn_wmma_f32_16x16x4_f32(
                false, a, false, b, (short)0, acc[mt], false, false);
        }
    }
#pragma unroll
    for (int mt = 0; mt < 5; ++mt)
#pragma unroll
        for (int r = 0; r < 8; ++r)    // C/D layout: VGPR r, group g -> M = r + 8g
            H[(size_t)(m0 + mt * 16 + r + 8 * lk) * F1 + n0 + lm] = acc[mt][r];
}

// ---------------- GEMM 2: h2 = x2 @ W2   [50000x512]x[512x64] ----------------

__global__ __launch_bounds__(128) void k_gemm2(const float* __restrict__ X,
                                               const float* __restrict__ W,
                                               float* __restrict__ H) {
    __shared__ float xs[16 * 516];                // 516 % 64 == 4, conflict-free
    const int m0  = blockIdx.x * 16;
    const int tid = threadIdx.x;
#ifdef USE_ASYNC_LDS
#pragma unroll
    for (int i = 0; i < 16; ++i) {                // 2048 float4 over 128 threads
        int idx = tid + i * 128;
        int row = idx >> 7;                       // 128 float4 per 512-float row
        int k4  = idx & 127;
        async_copy16(X + (size_t)(m0 + row) * F1 + k4 * 4, &xs[row * 516 + k4 * 4]);
    }
    __builtin_amdgcn_s_wait_asynccnt(0);
#else
    const float4* x4 = (const float4*)(X + (size_t)m0 * F1);
#pragma unroll
    for (int i = 0; i < 16; ++i) {
        int idx = tid + i * 128;
        int row = idx >> 7;
        int k4  = idx & 127;
        float4 v = x4[row * 128 + k4];
        float* dp = &xs[row * 516 + k4 * 4];
        dp[0] = v.x; dp[1] = v.y; dp[2] = v.z; dp[3] = v.w;
    }
#endif
    __syncthreads();

    const int lane = tid & 31;
    const int n0   = (tid >> 5) * 16;             // 4 waves -> N=64
    const int lm   = lane & 15;
    const int lk   = lane >> 4;
    v8f acc = {};
    for (int k0 = 0; k0 < F1; k0 += 4) {
        const int ka = k0 + 2 * lk;
        v2f a;
        a.x = xs[lm * 516 + ka];
        a.y = xs[lm * 516 + ka + 1];
        v2f b;
        b.x = W[(size_t)ka * FOUT + n0 + lm];
        b.y = W[(size_t)(ka + 1) * FOUT + n0 + lm];
        acc = __builtin_amdgcn_wmma_f32_16x16x4_f32(
            false, a, false, b, (short)0, acc, false, false);
    }
#pragma unroll
    for (int r = 0; r < 8; ++r)
        H[(size_t)(m0 + r + 8 * lk) * FOUT + n0 + lm] = acc[r];
}

// ---------------- attention coefficients ----------------

__global__ void k_alpha1(const float* __restrict__ H, const float* __restrict__ Asrc,
                         const float* __restrict__ Adst,
                         float* __restrict__ AS, float* __restrict__ AD) {
    int t = blockIdx.x * blockDim.x + threadIdx.x;
    if (t >= NNODES * H1N) return;
    int n = t >> 3, h = t & 7;
    const float4* hp = (const float4*)(H + (size_t)n * F1 + h * C1);
    const float4* sp = (const float4*)(Asrc + h * C1);
    const float4* dp = (const float4*)(Adst + h * C1);
    float ss = 0.f, sd = 0.f;
#pragma unroll
    for (int i = 0; i < 16; ++i) {
        float4 hv = hp[i], sv = sp[i], dv = dp[i];
        ss += hv.x * sv.x + hv.y * sv.y + hv.z * sv.z + hv.w * sv.w;
        sd += hv.x * dv.x + hv.y * dv.y + hv.z * dv.z + hv.w * dv.w;
    }
    AS[t] = ss; AD[t] = sd;
}

__global__ void k_alpha2(const float* __restrict__ H, const float* __restrict__ Asrc,
                         const float* __restrict__ Adst,
                         float* __restrict__ AS, float* __restrict__ AD) {
    int n = blockIdx.x * blockDim.x + threadIdx.x;
    if (n >= NNODES) return;
    const float4* hp = (const float4*)(H + (size_t)n * FOUT);
    const float4* sp = (const float4*)Asrc;
    const float4* dp = (const float4*)Adst;
    float ss = 0.f, sd = 0.f;
#pragma unroll
    for (int i = 0; i < 16; ++i) {
        float4 hv = hp[i], sv = sp[i], dv = dp[i];
        ss += hv.x * sv.x + hv.y * sv.y + hv.z * sv.z + hv.w * sv.w;
        sd += hv.x * dv.x + hv.y * dv.y + hv.z * dv.z + hv.w * dv.w;
    }
    AS[n] = ss; AD[n] = sd;
}

// ---------------- layer-1 edge sweeps (8 heads) ----------------

__global__ void k_edge1_max(const int* __restrict__ EI, const float* __restrict__ AS,
                            const float* __restrict__ AD, float* __restrict__ Ebuf,
                            float* __restrict__ MX) {
    int t = blockIdx.x * blockDim.x + threadIdx.x;
    if (t >= ETOT * H1N) return;
    int e = t >> 3, h = t & 7;
    int s, d; edge_sd(EI, e, s, d);
    float v = AS[s * H1N + h] + AD[d * H1N + h];
    v = v > 0.f ? v : NSLOPE * v;                 // leaky_relu
    Ebuf[t] = v;
    atomicMaxF(&MX[d * H1N + h], v);
}

__global__ void k_edge1_exp(const int* __restrict__ EI, float* __restrict__ Ebuf,
                            const float* __restrict__ MX, float* __restrict__ DN) {
    int t = blockIdx.x * blockDim.x + threadIdx.x;
    if (t >= ETOT * H1N) return;
    int e = t >> 3, h = t & 7;
    int s, d; edge_sd(EI, e, s, d); (void)s;
    float ex = __expf(Ebuf[t] - MX[d * H1N + h]);
    Ebuf[t] = ex;
    atomicAdd(&DN[d * H1N + h], ex);
}

// one wave per edge: coalesced gather of h1[src][0..511], coalesced atomic scatter
__global__ __launch_bounds__(256) void k_edge1_aggr(const int* __restrict__ EI,
                                                    const float* __restrict__ H,
                                                    const float* __restrict__ Ebuf,
                                                    const float* __restrict__ DN,
                                                    float* __restrict__ ACC) {
    int e = blockIdx.x * 8 + (threadIdx.x >> 5);
    if (e >= ETOT) return;
    int lane = threadIdx.x & 31;
    int s, d; edge_sd(EI, e, s, d);
    const float* hr   = H   + (size_t)s * F1;
    float*       outr = ACC + (size_t)d * F1;
#pragma unroll
    for (int h = 0; h < H1N; ++h) {
        float alpha = Ebuf[(size_t)e * H1N + h] / (DN[d * H1N + h] + 1e-16f);
#pragma unroll
        for (int j = 0; j < 2; ++j) {
            int c = h * C1 + j * 32 + lane;
            atomicAdd(&outr[c], alpha * hr[c]);
        }
    }
}

__global__ void k_elu_bias(float* __restrict__ A, const float* __restrict__ B) {
    int t = blockIdx.x * blockDim.x + threadIdx.x;
    if (t >= NNODES * F1) return;
    float v = A[t] + B[t & (F1 - 1)];
    A[t] = v > 0.f ? v : (__expf(v) - 1.f);       // ELU(alpha=1)
}

// ---------------- layer-2 edge sweeps (1 head) ----------------

__global__ void k_edge2_max(const int* __restrict__ EI, const float* __restrict__ AS,
                            const float* __restrict__ AD, float* __restrict__ Ebuf,
                            float* __restrict__ MX) {
    int e = blockIdx.x * blockDim.x + threadIdx.x;
    if (e >= ETOT) return;
    int s, d; edge_sd(EI, e, s, d);
    float v = AS[s] + AD[d];
    v = v > 0.f ? v : NSLOPE * v;
    Ebuf[e] = v;
    atomicMaxF(&MX[d], v);
}

__global__ void k_edge2_exp(const int* __restrict__ EI, float* __restrict__ Ebuf,
                            const float* __restrict__ MX, float* __restrict__ DN) {
    int e = blockIdx.x * blockDim.x + threadIdx.x;
    if (e >= ETOT) return;
    int s, d; edge_sd(EI, e, s, d); (void)s;
    float ex = __expf(Ebuf[e] - MX[d]);
    Ebuf[e] = ex;
    atomicAdd(&DN[d], ex);
}

__global__ __launch_bounds__(256) void k_edge2_aggr(const int* __restrict__ EI,
                                                    const float* __restrict__ H,
                                                    const float* __restrict__ Ebuf,
                                                    const float* __restrict__ DN,
                                                    float* __restrict__ ACC) {
    int e = blockIdx.x * 8 + (threadIdx.x >> 5);
    if (e >= ETOT) return;
    int lane = threadIdx.x & 31;
    int s, d; edge_sd(EI, e, s, d);
    float alpha = Ebuf[e] / (DN[d] + 1e-16f);
    const float* hr   = H   + (size_t)s * FOUT;
    float*       outr = ACC + (size_t)d * FOUT;
#pragma unroll
    for (int j = 0; j < 2; ++j) {
        int c = j * 32 + lane;
        atomicAdd(&outr[c], alpha * hr[c]);
    }
}

__global__ void k_final(const float* __restrict__ ACC, const float* __restrict__ B,
                        float* __restrict__ OUT) {
    int t = blockIdx.x * blockDim.x + threadIdx.x;
    if (t >= NNODES * FOUT) return;
    OUT[t] = ACC[t] + B[t & (FOUT - 1)];
}

// ---------------- launch ----------------

static inline int cdiv_i(long a, long b) { return (int)((a + b - 1) / b); }

extern "C" void kernel_launch(void* const* d_in, const int* in_sizes, int n_in,
                              void* d_out, int out_size, void* d_ws, size_t ws_size,
                              hipStream_t stream) {
    const float* X   = (const float*)d_in[0];
    const int*   EI  = (const int*)  d_in[1];   // [2, E] int32
    const float* W1  = (const float*)d_in[2];
    const float* As1 = (const float*)d_in[3];
    const float* Ad1 = (const float*)d_in[4];
    const float* B1  = (const float*)d_in[5];
    const float* W2  = (const float*)d_in[6];
    const float* As2 = (const float*)d_in[7];
    const float* Ad2 = (const float*)d_in[8];
    const float* B2  = (const float*)d_in[9];
    float* out = (float*)d_out;

    // workspace layout (floats)
    float* w = (float*)d_ws;
    size_t off = 0;
    float* H1b  = w + off; off += (size_t)NNODES * F1;    // h1
    float* ACC1 = w + off; off += (size_t)NNODES * F1;    // layer-1 accum -> x2 (in place)
    float* AS1  = w + off; off += (size_t)NNODES * H1N;
    float* AD1  = w + off; off += (size_t)NNODES * H1N;
    float* MX1  = w + off; off += (size_t)NNODES * H1N;
    float* DN1  = w + off; off += (size_t)NNODES * H1N;
    float* E1b  = w + off; off += (size_t)ETOT * H1N;     // e then ex (in place)
    float* H2b  = w + off; off += (size_t)NNODES * FOUT;
    float* AS2  = w + off; off += (size_t)NNODES;
    float* AD2  = w + off; off += (size_t)NNODES;
    float* MX2  = w + off; off += (size_t)NNODES;
    float* DN2  = w + off; off += (size_t)NNODES;
    float* E2b  = w + off; off += (size_t)ETOT;
    float* ACC2 = w + off; off += (size_t)NNODES * FOUT;

    // init (required every call: harness does not re-poison between replays)
    k_fill<<<cdiv_i(NNODES * H1N, 256), 256, 0, stream>>>(MX1, -INFINITY, NNODES * H1N);
    k_fill<<<cdiv_i(NNODES * H1N, 256), 256, 0, stream>>>(DN1, 0.f, NNODES * H1N);
    k_fill<<<cdiv_i((long)NNODES * F1, 256), 256, 0, stream>>>(ACC1, 0.f, NNODES * F1);
    k_fill<<<cdiv_i(NNODES, 256), 256, 0, stream>>>(MX2, -INFINITY, NNODES);
    k_fill<<<cdiv_i(NNODES, 256), 256, 0, stream>>>(DN2, 0.f, NNODES);
    k_fill<<<cdiv_i(NNODES * FOUT, 256), 256, 0, stream>>>(ACC2, 0.f, NNODES * FOUT);

    // ---- layer 1 ----
    k_gemm1<<<dim3(NNODES / 80, F1 / 128), 256, 0, stream>>>(X, W1, H1b);
    k_alpha1<<<cdiv_i(NNODES * H1N, 256), 256, 0, stream>>>(H1b, As1, Ad1, AS1, AD1);
    k_edge1_max<<<cdiv_i((long)ETOT * H1N, 256), 256, 0, stream>>>(EI, AS1, AD1, E1b, MX1);
    k_edge1_exp<<<cdiv_i((long)ETOT * H1N, 256), 256, 0, stream>>>(EI, E1b, MX1, DN1);
    k_edge1_aggr<<<cdiv_i(ETOT, 8), 256, 0, stream>>>(EI, H1b, E1b, DN1, ACC1);
    k_elu_bias<<<cdiv_i((long)NNODES * F1, 256), 256, 0, stream>>>(ACC1, B1);

    // ---- layer 2 ----
    k_gemm2<<<NNODES / 16, 128, 0, stream>>>(ACC1, W2, H2b);
    k_alpha2<<<cdiv_i(NNODES, 256), 256, 0, stream>>>(H2b, As2, Ad2, AS2, AD2);
    k_edge2_max<<<cdiv_i(ETOT, 256), 256, 0, stream>>>(EI, AS2, AD2, E2b, MX2);
    k_edge2_exp<<<cdiv_i(ETOT, 256), 256, 0, stream>>>(EI, E2b, MX2, DN2);
    k_edge2_aggr<<<cdiv_i(ETOT, 8), 256, 0, stream>>>(EI, H2b, E2b, DN2, ACC2);
    k_final<<<cdiv_i(NNODES * FOUT, 256), 256, 0, stream>>>(ACC2, B2, out);
}